// FusionAwareInterp_83786222010990
// MI455X (gfx1250) — compile-verified
//
#include <hip/hip_runtime.h>
#include <math.h>

// ---------------------------------------------------------------------------
// Problem constants (from reference)
// ---------------------------------------------------------------------------
#define BSZ   2
#define NPTS  8192
#define CH3D  64
#define HH    64
#define WW    160
#define QPIX  (HH * WW)        // 10240 pixels per batch
#define KNN3  3
#define MT_PER_B (QPIX / 16)   // 640 M-tiles per batch

// Workspace layout
// P2: per batch, NPTS/16 groups of 32 float2 (B-fragment order): 128 KB/batch
#define P_BYTES    (BSZ * NPTS * 2 * 8)         // 256 KB
#define KNN_BYTES  (BSZ * QPIX * KNN3 * 4)      // 240 KB
#define FINAL_OFF  (P_BYTES + KNN_BYTES)        // 16B aligned
#define FINAL_BYTES ((size_t)BSZ * QPIX * 64 * 4)

typedef __attribute__((ext_vector_type(2))) float v2f;
typedef __attribute__((ext_vector_type(8))) float v8f;

__device__ __forceinline__ unsigned umin32(unsigned a, unsigned b) { return a < b ? a : b; }
__device__ __forceinline__ unsigned umax32(unsigned a, unsigned b) { return a > b ? a : b; }

// Branchless keep-3-smallest of a key stream (k0<=k1<=k2): 5 min/max ops.
__device__ __forceinline__ void key3_insert(unsigned x,
                                            unsigned& k0, unsigned& k1, unsigned& k2) {
    unsigned nk2 = umin32(k2, umax32(k1, x));
    unsigned nk1 = umin32(k1, umax32(k0, x));
    unsigned nk0 = umin32(k0, x);
    k0 = nk0; k1 = nk1; k2 = nk2;
}

// ---------------------------------------------------------------------------
// Kernel 0: point table in WMMA B-fragment order.
// Group g (16 points): entries [g*32 + j]      = (-2u_j, -2v_j)   (K rows 0,1)
//                      entries [g*32 + 16 + j] = (u_j^2+v_j^2, 0) (K rows 2,3)
// => a 1024-point panel is a contiguous 16 KB slab, async-copyable to LDS,
//    and the KNN inner-loop B-load is one uniform ds_load_b64 per lane.
// ---------------------------------------------------------------------------
__global__ __launch_bounds__(256) void prep_points(const float* __restrict__ uv,
                                                   float2* __restrict__ P2) {
    int t = blockIdx.x * 256 + threadIdx.x;
    if (t >= BSZ * NPTS) return;
    int b = t / NPTS, n = t % NPTS;
    float u = uv[(b * 2 + 0) * NPTS + n];
    float v = uv[(b * 2 + 1) * NPTS + n];
    int g = n >> 4, j = n & 15;
    float2* base = P2 + (size_t)b * (NPTS * 2) + g * 32;
    base[j]      = make_float2(-2.f * u, -2.f * v);
    base[16 + j] = make_float2(u * u + v * v, 0.f);
}

// Issue one 16 KB panel copy: 256 threads x 4 x 16B async-to-LDS.
__device__ __forceinline__ void stage_panel(const char* gp, unsigned lbase, int tid) {
#pragma unroll
    for (int i = 0; i < 4; ++i) {
        unsigned off   = (unsigned)tid * 16u + (unsigned)i * 4096u;
        unsigned laddr = lbase + off;
        const char* ga = gp + off;
        asm volatile("global_load_async_to_lds_b128 %0, %1, off"
                     :: "v"(laddr), "v"(ga) : "memory");
    }
}

// ---------------------------------------------------------------------------
// Kernel 1: KNN. D = A x B + C with C preloaded per-row |g|^2 so the WMMA
// emits exact d^2 >= 0. Selection: (d2,idx) packed into one u32 key, kept as
// a running smallest-3 via a pure min/max network (no branches, no EXEC ops).
// Point panels are double-buffered through LDS with async-to-LDS copies
// overlapped against WMMA + selection (one barrier per panel).
// ---------------------------------------------------------------------------
__global__ __launch_bounds__(256) void knn_kernel(const float2* __restrict__ P2,
                                                  int* __restrict__ knn) {
    __shared__ __align__(16) float2 sp0[2048];    // 16 KB panel buffer A
    __shared__ __align__(16) float2 sp1[2048];    // 16 KB panel buffer B

    const int tid  = threadIdx.x;
    const int lane = tid & 31;
    const int wave = tid >> 5;
    const int t    = blockIdx.x * 8 + wave;       // 0..1279 (8 | 640 -> uniform b)
    const int b    = t / MT_PER_B;
    const int m0   = (t % MT_PER_B) * 16;
    const int lh   = lane & 15;

    // A fragment (16x4): lanes 0-15 hold K=0,1 = (gx,gy); lanes 16-31 hold K=2,3 = (1,0)
    v2f afrag;
    if (lane < 16) {
        int q = m0 + lane;
        afrag[0] = (float)(q % WW);
        afrag[1] = (float)(q / WW);
    } else {
        afrag[0] = 1.0f;
        afrag[1] = 0.0f;
    }

    // C preload: per-row |g|^2  ->  D = |g|^2 + |p|^2 - 2 g.p = d^2
    v8f cinit;
#pragma unroll
    for (int r = 0; r < 8; ++r) {
        int row  = m0 + r + ((lane < 16) ? 0 : 8);
        float gx = (float)(row % WW), gy = (float)(row / WW);
        cinit[r] = gx * gx + gy * gy;
    }

    unsigned k0[8], k1[8], k2[8];
#pragma unroll
    for (int r = 0; r < 8; ++r) k0[r] = k1[r] = k2[r] = 0xFFFFFFFFu;

    const char* gsrc = (const char*)(P2 + (size_t)b * (NPTS * 2));
    const unsigned lb0 = (unsigned)(size_t)&sp0[0];
    const unsigned lb1 = (unsigned)(size_t)&sp1[0];

    stage_panel(gsrc, lb0, tid);                  // prologue: panel 0 -> buf0

    for (int p = 0; p < NPTS / 1024; ++p) {
        asm volatile("s_wait_asynccnt 0x0" ::: "memory");
        __syncthreads();                          // panel p resident; prior reads done
        if (p < NPTS / 1024 - 1)                  // prefetch p+1 into the other buffer
            stage_panel(gsrc + (p + 1) * 16384, ((p + 1) & 1) ? lb1 : lb0, tid);

        const float2* cur = (p & 1) ? sp1 : sp0;
#pragma unroll 4
        for (int nt = 0; nt < 64; ++nt) {
            float2 bv = cur[nt * 32 + lane];      // one uniform ds_load_b64
            v2f bfrag;
            bfrag[0] = bv.x;
            bfrag[1] = bv.y;

            v8f d = __builtin_amdgcn_wmma_f32_16x16x4_f32(
                        false, afrag, false, bfrag, (short)0, cinit, false, false);

            unsigned n = (unsigned)(p * 1024 + nt * 16 + lh);
#pragma unroll
            for (int r = 0; r < 8; ++r) {
                // d2 >= 0: float bits are order-isomorphic as ints; a slightly
                // negative rounding result has the sign bit set -> smax with 0
                // clamps it in ONE v_max_i32 (replaces two v_max_num_f32).
                int bi = __float_as_int(d[r]);
                bi = bi < 0 ? 0 : bi;
                unsigned key = ((unsigned)bi & 0xFFFFE000u) | n;  // v_and_or_b32
                key3_insert(key, k0[r], k1[r], k2[r]);
            }
        }
    }

    // Cross-lane merge inside each 16-lane half (rows m0+r / m0+8+r)
#pragma unroll
    for (int r = 0; r < 8; ++r) {
#pragma unroll
        for (int m = 1; m < 16; m <<= 1) {
            unsigned o0 = (unsigned)__shfl_xor((int)k0[r], m);
            unsigned o1 = (unsigned)__shfl_xor((int)k1[r], m);
            unsigned o2 = (unsigned)__shfl_xor((int)k2[r], m);
            key3_insert(o0, k0[r], k1[r], k2[r]);
            key3_insert(o1, k0[r], k1[r], k2[r]);
            key3_insert(o2, k0[r], k1[r], k2[r]);
        }
    }

    // Writers: lane r (half 0) -> row m0+r, lane 16+r (half 1) -> row m0+8+r
#pragma unroll
    for (int r = 0; r < 8; ++r) {
        if (lh == r) {
            int row  = m0 + r + ((lane >= 16) ? 8 : 0);
            int base = (b * QPIX + row) * KNN3;
            knn[base + 0] = (int)(k0[r] & 0x1FFFu);
            knn[base + 1] = (int)(k1[r] & 0x1FFFu);
            knn[base + 2] = (int)(k2[r] & 0x1FFFu);
        }
    }
}

// ---------------------------------------------------------------------------
// Kernel 2: gather knn uv/feat_3d, tiny MLP (3->16 relu, 16->64 sigmoid),
// weighted sum over k. One thread per pixel; weights staged in LDS.
// Writes final as [b][q][64] (row-major channels) for the WMMA A-loads.
// ---------------------------------------------------------------------------
__global__ __launch_bounds__(256) void interp_kernel(
        const float* __restrict__ uv, const float* __restrict__ f3d,
        const float* __restrict__ w1, const float* __restrict__ b1,
        const float* __restrict__ w2, const float* __restrict__ b2,
        const int* __restrict__ knn, float* __restrict__ finalbuf) {
    __shared__ float s_w1[48], s_b1[16], s_w2[1024], s_b2[64];
    const int tid = threadIdx.x;
    if (tid < 48) s_w1[tid] = w1[tid];
    if (tid < 16) s_b1[tid] = b1[tid];
    for (int i = tid; i < 1024; i += 256) s_w2[i] = w2[i];
    if (tid < 64) s_b2[tid] = b2[tid];
    __syncthreads();

    const int p = blockIdx.x * 256 + tid;      // 0..20479 exactly
    const int b = p / QPIX, q = p % QPIX;
    const float fx = (float)(q % WW);
    const float fy = (float)(q / WW);

    float acc[64];
#pragma unroll
    for (int c = 0; c < 64; ++c) acc[c] = 0.f;

    const int base = (b * QPIX + q) * KNN3;
#pragma unroll 1
    for (int k = 0; k < KNN3; ++k) {
        int n   = knn[base + k];
        float u = uv[(b * 2 + 0) * NPTS + n];
        float v = uv[(b * 2 + 1) * NPTS + n];
        float ox = u - fx, oy = v - fy;
        float on = sqrtf(ox * ox + oy * oy);
        float h1[16];
#pragma unroll
        for (int j = 0; j < 16; ++j) {
            float z = s_w1[j * 3 + 0] * ox + s_w1[j * 3 + 1] * oy +
                      s_w1[j * 3 + 2] * on + s_b1[j];
            h1[j] = fmaxf(z, 0.f);
        }
        const float* f3 = f3d + (size_t)b * CH3D * NPTS + n;
#pragma unroll
        for (int c = 0; c < 64; ++c) {
            float z = s_b2[c];
#pragma unroll
            for (int j = 0; j < 16; ++j) z += s_w2[c * 16 + j] * h1[j];
            float s = 1.f / (1.f + __expf(-z));
            acc[c] += s * f3[(size_t)c * NPTS];
        }
    }

    float4* outp = (float4*)(finalbuf + (size_t)(b * QPIX + q) * 64);
#pragma unroll
    for (int c = 0; c < 64; c += 4)
        outp[c >> 2] = make_float4(acc[c], acc[c + 1], acc[c + 2], acc[c + 3]);
}

// ---------------------------------------------------------------------------
// Kernel 3: out = relu(w3 x final + b3) as WMMA GEMM.
// M = pixels (16/tile), N = 64 outputs (4 tiles), K = 64 (16 chained x4 steps).
// One wave per (M-tile, N-tile); w3/b3 in LDS; epilogue transposed through
// padded LDS (stride 20: conflict-free, 16B-aligned) for coalesced b128 stores.
// ---------------------------------------------------------------------------
__global__ __launch_bounds__(256) void out_gemm(const float* __restrict__ finalbuf,
                                                const float* __restrict__ w3,
                                                const float* __restrict__ b3,
                                                float* __restrict__ out) {
    __shared__ __align__(16) float s_w3[4096];
    __shared__ float s_b3[64];
    __shared__ __align__(16) float stile[8 * 16 * 20];   // 8 waves x (16 o x 20 pad)
    const int tid = threadIdx.x;
    for (int i = tid; i < 4096; i += 256) s_w3[i] = w3[i];
    if (tid < 64) s_b3[tid] = b3[tid];
    __syncthreads();

    const int lane = tid & 31, wave = tid >> 5;
    const int j  = blockIdx.x * 8 + wave;         // 0..5119
    const int mt = j >> 2, nt = j & 3;
    const int b  = mt / MT_PER_B;
    const int m0 = (mt % MT_PER_B) * 16;
    const int o0 = nt * 16;
    const int lh = lane & 15;

    const int q  = m0 + lh;                       // A row for this lane
    const int cA = (lane < 16) ? 0 : 2;           // K sub-row per half-wave
    const float2* arow = (const float2*)(finalbuf + (size_t)(b * QPIX + q) * 64 + cA);
    const float2* brow = (const float2*)(s_w3 + (o0 + lh) * 64 + cA);

    v8f acc = {};
#pragma unroll
    for (int k0 = 0; k0 < 64; k0 += 4) {
        float2 av = arow[k0 >> 1];                // (k0+cA, k0+cA+1)
        float2 bv = brow[k0 >> 1];
        v2f a, bb;
        a[0] = av.x;  a[1] = av.y;
        bb[0] = bv.x; bb[1] = bv.y;
        acc = __builtin_amdgcn_wmma_f32_16x16x4_f32(
                  false, a, false, bb, (short)0, acc, false, false);
    }

    // bias + relu, stage tile transposed in LDS: stile[w][o_loc][m_loc]
    const float bias   = s_b3[o0 + lh];
    const int   mshift = (lane < 16) ? 0 : 8;
    float* st = stile + wave * 320;
#pragma unroll
    for (int r = 0; r < 8; ++r) {
        float v = fmaxf(acc[r] + bias, 0.f);
        st[lh * 20 + r + mshift] = v;
    }
    __syncthreads();

    // Coalesced writeback: 8 tiles x 16 o-rows x 4 float4 = 512 float4 per block.
#pragma unroll
    for (int id = tid; id < 512; id += 256) {
        int w     = id >> 6;
        int rem   = id & 63;
        int o_loc = rem >> 2;
        int seg   = rem & 3;
        int jw  = blockIdx.x * 8 + w;
        int mtw = jw >> 2, ntw = jw & 3;
        int bw  = mtw / MT_PER_B;
        int m0w = (mtw % MT_PER_B) * 16;
        int o0w = ntw * 16;
        float4 vv = *(const float4*)(stile + w * 320 + o_loc * 20 + seg * 4);
        *(float4*)(out + ((size_t)(bw * 64 + o0w + o_loc)) * QPIX + m0w + seg * 4) = vv;
    }
}

// ---------------------------------------------------------------------------
extern "C" void kernel_launch(void* const* d_in, const int* in_sizes, int n_in,
                              void* d_out, int out_size, void* d_ws, size_t ws_size,
                              hipStream_t stream) {
    const float* uv  = (const float*)d_in[0];
    // d_in[1] (feat_2d) is unused by the reference computation (shape only)
    const float* f3d = (const float*)d_in[2];
    const float* w1  = (const float*)d_in[3];
    const float* b1  = (const float*)d_in[4];
    const float* w2  = (const float*)d_in[5];
    const float* b2  = (const float*)d_in[6];
    const float* w3  = (const float*)d_in[7];
    const float* b3  = (const float*)d_in[8];
    float* out = (float*)d_out;

    if (ws_size < FINAL_OFF + FINAL_BYTES) return;  // deterministic no-op guard

    char*   wsb      = (char*)d_ws;
    float2* P2       = (float2*)wsb;
    int*    knn      = (int*)(wsb + P_BYTES);
    float*  finalbuf = (float*)(wsb + FINAL_OFF);

    prep_points  <<<(BSZ * NPTS + 255) / 256, 256, 0, stream>>>(uv, P2);
    knn_kernel   <<<(BSZ * MT_PER_B) / 8,     256, 0, stream>>>(P2, knn);
    interp_kernel<<<(BSZ * QPIX) / 256,       256, 0, stream>>>(uv, f3d, w1, b1, w2, b2,
                                                                knn, finalbuf);
    out_gemm     <<<(BSZ * MT_PER_B * 4) / 8, 256, 0, stream>>>(finalbuf, w3, b3, out);
}